// KernelApply_42709154791905
// MI455X (gfx1250) — compile-verified
//
#include <hip/hip_runtime.h>
#include <type_traits>

namespace {
constexpr int Hh = 128, Ww = 128, Kk = 21, KKt = 441, Cc = 10;
constexpr int PLANE = Hh * Ww;
}

__global__ __launch_bounds__(128, 1) void kpn_splat_softmax_kernel(
    const float* __restrict__ data,
    const float* __restrict__ kernels,
    float* __restrict__ out)
{
    // Per-wave double-buffered staging: 21 kernel taps x 32 lanes, and a
    // 64-wide x 3-channel data window. 4 waves/block -> 27.6 KB LDS.
    __shared__ float sK[4][2][Kk][32];
    __shared__ float sD[4][2][3][64];

    const int tid  = threadIdx.x;
    const int lane = tid & 31;
    const int wid  = tid >> 5;
    const int x    = tid;          // 0..127 (full row per block)
    const int x0   = x - lane;     // wave strip base
    const int b    = blockIdx.x >> 7;
    const int y    = blockIdx.x & (Hh - 1);

    // Valid tap ranges (i-range is wave-uniform; j validity is per-lane)
    const int i0  = (y < Cc) ? (Cc - y) : 0;
    const int i1  = (y > Hh - 1 - Cc) ? (Hh + Cc - y) : Kk;
    const int jlo = (x < Cc) ? (Cc - x) : 0;
    const int jhi = (x > Ww - 1 - Cc) ? (Ww + Cc - x) : Kk;
    const int nvalid = (i1 - i0) * (jhi - jlo);

    // OOB taps each contribute exp(0)=1 to the softmax denominator only.
    const float s0 = (float)(KKt - nvalid);
    float s  = s0;
    float a0 = 0.f, a1 = 0.f, a2 = 0.f;

    const float* kb = kernels + (size_t)b * KKt * PLANE;
    const float* db = data    + (size_t)b * 3   * PLANE;

    // wave-uniform: strips 1 and 2 need no per-lane x masking
    const bool fullx = (x0 >= Cc) && (x0 + 31 <= Ww - 1 - Cc);

    auto stage = [&](int i, int buf) {
        // ensure all outstanding DS reads from the buffer we overwrite are done
        asm volatile("s_wait_dscnt 0" ::: "memory");
        const int yy = y + i - Cc;   // in [0,Hh) for i in [i0,i1)
        // tap (i,j) lives in plane ch = (20-i)*21 + (20-j)
        const float* krow = kb + (size_t)((Kk - 1 - i) * Kk + (Kk - 1)) * PLANE
                               + (size_t)yy * Ww;
        unsigned lK = (unsigned)(size_t)(&sK[wid][buf][0][lane]);
#pragma unroll
        for (int j = 0; j < Kk; ++j) {
            int xx  = x + j - Cc;
            int xxc = xx < 0 ? 0 : (xx > Ww - 1 ? Ww - 1 : xx); // clamped; masked later
            const float* gp = krow - (size_t)j * PLANE + xxc;
            unsigned la = lK + (unsigned)(j * 32 * 4);
            asm volatile("global_load_async_to_lds_b32 %0, %1, off"
                         :: "v"(la), "v"(gp) : "memory");
        }
        // data window: columns [x0-10, x0+54) per channel, 2 B32 per lane
#pragma unroll
        for (int c = 0; c < 3; ++c) {
            const float* drow = db + (size_t)c * PLANE + (size_t)yy * Ww;
            int c0  = x0 - Cc + lane;
            int c0c = c0 < 0 ? 0 : (c0 > Ww - 1 ? Ww - 1 : c0);
            int c1  = c0 + 32;
            int c1c = c1 > Ww - 1 ? Ww - 1 : c1;
            unsigned lD = (unsigned)(size_t)(&sD[wid][buf][c][lane]);
            asm volatile("global_load_async_to_lds_b32 %0, %1, off"
                         :: "v"(lD), "v"(drow + c0c) : "memory");
            asm volatile("global_load_async_to_lds_b32 %0, %1, off"
                         :: "v"(lD + 128u), "v"(drow + c1c) : "memory");
        }
    };

    auto run = [&](auto full_c) {
        constexpr bool FULL = decltype(full_c)::value;
        stage(i0, 0);
        for (int i = i0; i < i1; ++i) {
            const int cur = (i - i0) & 1;
            if (i + 1 < i1) {
                stage(i + 1, cur ^ 1);                 // prefetch next row
                asm volatile("s_wait_asynccnt 27" ::: "memory"); // row i done
            } else {
                asm volatile("s_wait_asynccnt 0" ::: "memory");
            }
#pragma unroll
            for (int j = 0; j < Kk; ++j) {
                float p = __expf(sK[wid][cur][j][lane]);
                if (!FULL) {
                    int xx = x + j - Cc;
                    if ((unsigned)xx >= (unsigned)Ww) p = 0.0f;
                }
                float d0 = sD[wid][cur][0][lane + j];
                float d1 = sD[wid][cur][1][lane + j];
                float d2 = sD[wid][cur][2][lane + j];
                s  += p;
                a0 = fmaf(p, d0, a0);
                a1 = fmaf(p, d1, a1);
                a2 = fmaf(p, d2, a2);
            }
        }
    };
    if (fullx) run(std::integral_constant<bool, true>{});
    else       run(std::integral_constant<bool, false>{});

    // sum of valid softmax weights: every valid tap added p to s exactly once,
    // and the OOB taps contributed the constant s0 => av = s - s0.
    const float av  = s - s0;
    const float inv = 1.0f / s;
    const int pix = y * Ww + x;
    out[(size_t)(b * 3 + 0) * PLANE + pix] = a0 * inv;
    out[(size_t)(b * 3 + 1) * PLANE + pix] = a1 * inv;
    out[(size_t)(b * 3 + 2) * PLANE + pix] = a2 * inv;
    out[(size_t)(4 * 3) * PLANE + (size_t)b * PLANE + pix] = av * inv; // sum_w
}

extern "C" void kernel_launch(void* const* d_in, const int* in_sizes, int n_in,
                              void* d_out, int out_size, void* d_ws, size_t ws_size,
                              hipStream_t stream) {
    (void)in_sizes; (void)n_in; (void)out_size; (void)d_ws; (void)ws_size;
    const float* data    = (const float*)d_in[0];
    const float* kernels = (const float*)d_in[1];
    float* out = (float*)d_out;
    // 4 batches x 128 rows; 128 threads = 4 wave32 strips per row
    kpn_splat_softmax_kernel<<<dim3(4 * Hh), dim3(128), 0, stream>>>(data, kernels, out);
}